// HLayer_2379411882433
// MI455X (gfx1250) — compile-verified
//
#include <hip/hip_runtime.h>
#include <stdint.h>

// Hadamard on qubit 5 of a 24-qubit f32 state, batch 2.
// Pure streaming butterfly: 268 MB traffic, 67 MFLOP -> HBM-bound (~11.5 us
// at 23.3 TB/s). Data movement via CDNA5 Tensor Data Mover (2x2048 strided
// pair-tiles into LDS, TENSORcnt-pipelined), butterfly in VALU, NT B128 stores.

typedef unsigned int u32x4 __attribute__((ext_vector_type(4)));
typedef int          i32x8 __attribute__((ext_vector_type(8)));
typedef int          i32x4 __attribute__((ext_vector_type(4)));
typedef float        f32x4 __attribute__((ext_vector_type(4)));

static constexpr uint32_t kR      = 1u << 18;          // 2^(24-5-1) cols per row
static constexpr uint32_t kRows   = 128;               // 2 batch * 32 L * 2 j
static constexpr uint32_t kPairs  = kRows / 2;         // 64 row pairs
static constexpr uint32_t kGroups = 32;                // column slabs per pair
static constexpr uint32_t kW      = 2048;              // tile width (floats)
static constexpr uint32_t kTiles  = (kR / kGroups) / kW; // 4 tiles per block

__global__ __launch_bounds__(32)
void hadamard_q5_tdm(const float* __restrict__ in, float* __restrict__ out) {
  // [buf][row j][col] : 2 * 2 * 2048 * 4B = 32 KB (double buffer)
  __shared__ float lds[2][2][kW];

  const uint32_t pair = blockIdx.x >> 5;               // 0..63
  const uint32_t grp  = blockIdx.x & (kGroups - 1u);   // 0..31
  // element offset of this block's first (j=0) column chunk
  const uint64_t base = (uint64_t)pair * (2ull * kR) + (uint64_t)grp * (kR / kGroups);

  // LDS byte offset: low 32 bits of the shared-aperture flat address are the
  // LDS offset.
  const uint32_t lds_base = (uint32_t)(uintptr_t)(&lds[0][0][0]);

  auto tdm_issue = [&](uint32_t t) {
    const uint64_t gaddr = (uint64_t)(uintptr_t)(in + base + (uint64_t)t * kW);
    // ---- D# group 0 (128b): count=1 | lds_addr | global_addr[56:0] | type=2
    u32x4 d0;
    d0.x = 1u;                                              // count=1, user mode
    d0.y = lds_base + (t & 1u) * (2u * kW * 4u);            // dest LDS buffer
    d0.z = (uint32_t)gaddr;                                 // global_addr[31:0]
    d0.w = ((uint32_t)(gaddr >> 32) & 0x01FFFFFFu)          // global_addr[56:32]
         | (2u << 30);                                      // type = 2 ("image")
    // ---- D# group 1 (256b): 2D tensor kRows x kR, tile 2 x kW
    i32x8 d1;
    d1[0] = (int)(2u << 16);                    // wg_mask=0, data_size=2 (4B)
    d1[1] = (int)((kR & 0xFFFFu) << 16);        // tensor_dim0[15:0] @ bits 63:48
    d1[2] = (int)((kR >> 16) | (kRows << 16));  // tensor_dim0[31:16] | tensor_dim1[15:0]
    d1[3] = (int)(kW << 16);                    // tensor_dim1[31:16]=0 | tile_dim0
    d1[4] = 2;                                  // tile_dim1=2, tile_dim2=0 (2D)
    d1[5] = (int)kR;                            // tensor_dim0_stride[31:0]
    d1[6] = 0;                                  // stride0[47:32] | stride1[15:0]
    d1[7] = 0;                                  // stride1[47:16]
    const i32x4 z4 = {0, 0, 0, 0};              // groups 2/3 unused (2D tensor)
    const i32x8 z8 = {0, 0, 0, 0, 0, 0, 0, 0};  // 6-arg toolchain extra group
    __builtin_amdgcn_tensor_load_to_lds(d0, d1, z4, z4, z8, 0);
  };

  tdm_issue(0);
  for (uint32_t t = 0; t < kTiles; ++t) {
    if (t + 1u < kTiles) {
      tdm_issue(t + 1u);                        // keep 2 tiles in flight
      __builtin_amdgcn_s_wait_tensorcnt(1);     // in-order: oldest (tile t) done
    } else {
      __builtin_amdgcn_s_wait_tensorcnt(0);
    }

    const uint32_t buf = t & 1u;
    const float s = 0.70710678118654752440f;    // 1/sqrt(2)
    float* o0 = out + base + (uint64_t)t * kW;  // j = 0 row chunk
    float* o1 = o0 + kR;                        // j = 1 row chunk

#pragma unroll
    for (uint32_t it = 0; it < kW / (32u * 4u); ++it) {  // 16 f32x4 per lane
      const uint32_t idx = (it * 32u + threadIdx.x) * 4u;
      f32x4 a = *(const f32x4*)&lds[buf][0][idx];        // ds_load_b128
      f32x4 b = *(const f32x4*)&lds[buf][1][idx];
      f32x4 y0 = (a + b) * s;
      f32x4 y1 = (a - b) * s;
      __builtin_nontemporal_store(y0, (f32x4*)(o0 + idx)); // stream, don't rinse L2
      __builtin_nontemporal_store(y1, (f32x4*)(o1 + idx));
    }
  }
}

extern "C" void kernel_launch(void* const* d_in, const int* in_sizes, int n_in,
                              void* d_out, int out_size, void* d_ws, size_t ws_size,
                              hipStream_t stream) {
  (void)in_sizes; (void)n_in; (void)d_ws; (void)ws_size; (void)out_size;
  const float* in = (const float*)d_in[0];
  float* out = (float*)d_out;
  // 64 pairs * 32 column slabs = 2048 single-wave workgroups; 32 KB LDS each
  // -> up to 10 resident per WGP against the 320 KB pool; TDM hides latency.
  hipLaunchKernelGGL(hadamard_q5_tdm, dim3(kPairs * kGroups), dim3(32), 0, stream,
                     in, out);
}